// MetaGraphLearner_90666759618969
// MI455X (gfx1250) — compile-verified
//
#include <hip/hip_runtime.h>

typedef __attribute__((ext_vector_type(2))) float v2f;
typedef __attribute__((ext_vector_type(8))) float v8f;

#define B_  2
#define N_  1024
#define H_  128
#define E_  32
#define M_  (B_ * N_)   // 2048 rows
#define EC  64          // combined columns: [hi(32) | hj+b1(32)]

// workspace layout in floats
#define WS_WC2    0              // 64 x 128
#define WS_BIAS   (64 * 128)     // 64
#define WS_HIALL  8320           // 2048 x 64 (256B aligned start)

// ---------------------------------------------------------------------------
// Kernel 1: fold the GEMM chain.
// Wc2[e'][h] = sum_o W1{i|j}[e][o] * Wt[o][h]   (e'=0..63, rows 32..63 use W1j)
// bias[e']   = sum_o W1{i|j}[e][o] * bt[o]  (+ b1[e] on the j side)
// ---------------------------------------------------------------------------
__global__ void prep_kernel(const float* __restrict__ Wt, const float* __restrict__ bt,
                            const float* __restrict__ W1, const float* __restrict__ b1,
                            float* __restrict__ ws) {
    int idx = blockIdx.x * blockDim.x + threadIdx.x;   // 0..8191
    int ep  = idx >> 7;                                 // 0..63
    int h   = idx & 127;
    int e   = ep & 31;
    int isj = ep >> 5;
    const float* w1row = W1 + e * (2 * H_) + isj * H_;

    float acc = 0.f;
    #pragma unroll 8
    for (int o = 0; o < H_; ++o)
        acc = fmaf(w1row[o], Wt[o * H_ + h], acc);
    ws[WS_WC2 + ep * H_ + h] = acc;

    if (h == 0) {
        float bacc = isj ? b1[e] : 0.f;
        #pragma unroll 8
        for (int o = 0; o < H_; ++o)
            bacc = fmaf(w1row[o], bt[o], bacc);
        ws[WS_BIAS + ep] = bacc;
    }
}

// ---------------------------------------------------------------------------
// Kernel 2: fp32 WMMA GEMM.  hi_all(2048x64) = nf(2048x128) @ Wc2^T + bias
// One wave per 16x16 tile; V_WMMA_F32_16X16X4_F32, 32 MACs over K=128.
// A lane layout (ISA 7.12.2): lane L holds row (L&15), K pair starting at
// k0 + 2*(L>=16) -> contiguous float2.  B[k][n] = Wc2[n][k] mirrors it.
// C layout: VGPR j, lane L -> row = j + 8*(L>=16), col = L&15.
// ---------------------------------------------------------------------------
__global__ void gemm_wmma_kernel(const float* __restrict__ nf,
                                 const float* __restrict__ wc2,
                                 const float* __restrict__ bias,
                                 float* __restrict__ hi_all) {
    int tile = blockIdx.x;          // 512 tiles = 128 (M) x 4 (N)
    int tm   = tile >> 2;
    int tn   = tile & 3;
    int lane = threadIdx.x;
    int rc   = lane & 15;
    int khi  = lane >> 4;           // 0 or 1

    const float* arow = nf  + (size_t)(tm * 16 + rc) * H_ + 2 * khi;
    const float* brow = wc2 + (size_t)(tn * 16 + rc) * H_ + 2 * khi;

    v8f c = {0.f, 0.f, 0.f, 0.f, 0.f, 0.f, 0.f, 0.f};
    #pragma unroll 4
    for (int k0 = 0; k0 < H_; k0 += 4) {
        v2f a = *(const v2f*)(arow + k0);
        v2f b = *(const v2f*)(brow + k0);
        c = __builtin_amdgcn_wmma_f32_16x16x4_f32(false, a, false, b,
                                                  (short)0, c, false, false);
    }

    float bv = bias[tn * 16 + rc];
    float* outp = hi_all + (size_t)(tm * 16 + 8 * khi) * EC + tn * 16 + rc;
    #pragma unroll
    for (int j = 0; j < 8; ++j)
        outp[(size_t)j * EC] = c[j] + bv;
}

// ---------------------------------------------------------------------------
// Kernel 3: pairwise edge scores (the hot loop).
// out[b,n,m] = sigmoid( sum_e relu(hi[n,e] + hjb[m,e]) * W2[e] + b2 ) * adj[n,m]
// 64x64 tile per 256-thread block; LDS tiles stored transposed [e][n] so the
// inner loop is 2x ds_load_b128 + broadcast per e, feeding 16 add/max/fma.
// ---------------------------------------------------------------------------
__global__ void pairwise_kernel(const float* __restrict__ hi_all,
                                const float* __restrict__ adj,
                                const float* __restrict__ W2,
                                const float* __restrict__ b2,
                                float* __restrict__ out) {
    __shared__ float hiT[E_ * 64];
    __shared__ float hjT[E_ * 64];
    __shared__ float w2s[E_];

    int t  = threadIdx.x;
    int m0 = blockIdx.x * 64;
    int n0 = blockIdx.y * 64;
    int bz = blockIdx.z;
    const float* base = hi_all + (size_t)bz * N_ * EC;

    #pragma unroll
    for (int i = 0; i < 8; ++i) {       // 2048 elements per tile side
        int idx = t * 8 + i;
        int r = idx >> 5;
        int e = idx & 31;
        hiT[e * 64 + r] = base[(size_t)(n0 + r) * EC + e];
        hjT[e * 64 + r] = base[(size_t)(m0 + r) * EC + 32 + e];
    }
    if (t < E_) w2s[t] = W2[t];
    __syncthreads();

    int tx = t & 15;                    // 4 m-columns
    int ty = t >> 4;                    // 4 n-rows
    float acc[16];
    #pragma unroll
    for (int i = 0; i < 16; ++i) acc[i] = 0.f;

    #pragma unroll 4
    for (int e = 0; e < E_; ++e) {
        float4 hq = *(const float4*)&hiT[e * 64 + ty * 4];
        float4 wq = *(const float4*)&hjT[e * 64 + tx * 4];
        float w = w2s[e];
        float hv[4] = {hq.x, hq.y, hq.z, hq.w};
        float wv[4] = {wq.x, wq.y, wq.z, wq.w};
        #pragma unroll
        for (int i = 0; i < 4; ++i)
            #pragma unroll
            for (int j = 0; j < 4; ++j)
                acc[i * 4 + j] = fmaf(fmaxf(hv[i] + wv[j], 0.f), w, acc[i * 4 + j]);
    }

    float b2v = b2[0];
    #pragma unroll
    for (int i = 0; i < 4; ++i) {
        int n = n0 + ty * 4 + i;
        float4 a4 = *(const float4*)&adj[(size_t)n * N_ + m0 + tx * 4];
        float4 o4;
        o4.x = a4.x / (1.f + __expf(-(acc[i * 4 + 0] + b2v)));
        o4.y = a4.y / (1.f + __expf(-(acc[i * 4 + 1] + b2v)));
        o4.z = a4.z / (1.f + __expf(-(acc[i * 4 + 2] + b2v)));
        o4.w = a4.w / (1.f + __expf(-(acc[i * 4 + 3] + b2v)));
        *(float4*)&out[((size_t)bz * N_ + n) * N_ + m0 + tx * 4] = o4;
    }
}

// ---------------------------------------------------------------------------
extern "C" void kernel_launch(void* const* d_in, const int* in_sizes, int n_in,
                              void* d_out, int out_size, void* d_ws, size_t ws_size,
                              hipStream_t stream) {
    const float* nf  = (const float*)d_in[0];   // (B,N,H)
    const float* adj = (const float*)d_in[1];   // (N,N)
    const float* Wt  = (const float*)d_in[2];   // (H,H)
    const float* bt  = (const float*)d_in[3];   // (H,)
    const float* W1  = (const float*)d_in[4];   // (E,2H)
    const float* b1  = (const float*)d_in[5];   // (E,)
    const float* W2  = (const float*)d_in[6];   // (1,E)
    const float* b2  = (const float*)d_in[7];   // (1,)
    float* ws  = (float*)d_ws;
    float* out = (float*)d_out;

    prep_kernel<<<32, 256, 0, stream>>>(Wt, bt, W1, b1, ws);
    gemm_wmma_kernel<<<512, 32, 0, stream>>>(nf, ws + WS_WC2, ws + WS_BIAS,
                                             ws + WS_HIALL);
    pairwise_kernel<<<dim3(N_ / 64, N_ / 64, B_), 256, 0, stream>>>(
        ws + WS_HIALL, adj, W2, b2, out);
}